// GAT_25812753449275
// MI455X (gfx1250) — compile-verified
//
#include <hip/hip_runtime.h>
#include <cstdint>

typedef __attribute__((ext_vector_type(2))) float v2f;
typedef __attribute__((ext_vector_type(8))) float v8f;

#define NEG_SLOPE 0.2f

__device__ __forceinline__ float lrelu(float t) { return t > 0.f ? t : NEG_SLOPE * t; }

// Bit-trick float atomic max: works for mixed signs, init = -inf.
__device__ __forceinline__ void atomicMaxF(float* addr, float v) {
  if (v >= 0.f) atomicMax((int*)addr, __float_as_int(v));
  else          atomicMin((unsigned int*)addr, (unsigned int)__float_as_int(v));
}

__global__ void gat_fill(float* __restrict__ p, float v, int n) {
  int i = blockIdx.x * blockDim.x + threadIdx.x;
  if (i < n) p[i] = v;
}

// ---------------- Layer 1 GEMM: h[N,64] = x[N,512] @ W1[512,64] (f32 WMMA) ---
__global__ void gat_gemm1(const float* __restrict__ x, const float* __restrict__ W,
                          float* __restrict__ h, int Nn) {
  int wave = (int)((blockIdx.x * blockDim.x + threadIdx.x) >> 5);
  int lane = threadIdx.x & 31;
  int tiles = (Nn >> 4) * 4;                 // (N/16) row tiles x 4 col tiles
  if (wave >= tiles) return;                 // wave-uniform: EXEC stays all-ones
  int mt = wave >> 2, nt = wave & 3;
  int row   = mt * 16 + (lane & 15);
  int col   = nt * 16 + (lane & 15);
  int khalf = (lane >> 4) * 2;               // lanes 0-15 -> K 0..1, 16-31 -> K 2..3
  const float* xr = x + (size_t)row * 512;
  v8f acc = {};
  for (int k0 = 0; k0 < 512; k0 += 4) {
    v2f a, b;
    a.x = xr[k0 + khalf];
    a.y = xr[k0 + khalf + 1];
    b.x = W[(k0 + khalf) * 64 + col];
    b.y = W[(k0 + khalf + 1) * 64 + col];
    acc = __builtin_amdgcn_wmma_f32_16x16x4_f32(false, a, false, b, (short)0, acc,
                                                false, false);
  }
  int rbase = mt * 16 + ((lane & 16) ? 8 : 0);   // C/D: VGPR j -> rows j, j+8
#pragma unroll
  for (int j = 0; j < 8; ++j)
    h[(size_t)(rbase + j) * 64 + col] = acc[j];
}

// ---------------- alpha_src/alpha_dst for layer 1: per (node,head) -----------
__global__ void gat_alpha1(const float* __restrict__ h, const float* __restrict__ a_src,
                           const float* __restrict__ a_dst, float* __restrict__ as,
                           float* __restrict__ ad, int NH) {
  int i = blockIdx.x * blockDim.x + threadIdx.x;
  if (i >= NH) return;
  int head = i & 7;
  const float* hp = h + (size_t)i * 8;      // h is [n][head][8] contiguous
  const float* sp = a_src + head * 8;
  const float* dp = a_dst + head * 8;
  float vs = 0.f, vd = 0.f;
#pragma unroll
  for (int f = 0; f < 8; ++f) { float hv = hp[f]; vs += hv * sp[f]; vd += hv * dp[f]; }
  as[i] = vs; ad[i] = vd;
}

// ---------------- edge pass A (layer 1): segment max per (dst,head) ----------
__global__ void gat_emax1(const int* __restrict__ src, const int* __restrict__ dst,
                          const float* __restrict__ as, const float* __restrict__ ad,
                          float* __restrict__ m, int E, int N) {
  int i = blockIdx.x * blockDim.x + threadIdx.x;
  int tot = (E + N) * 8;
  if (i >= tot) return;
  int e = i >> 3, hh = i & 7;
  int s_ = (e < E) ? src[e] : (e - E);      // self-loop synthesis
  int d_ = (e < E) ? dst[e] : (e - E);
  float t = lrelu(as[s_ * 8 + hh] + ad[d_ * 8 + hh]);
  atomicMaxF(&m[d_ * 8 + hh], t);
}

// ---------------- edge pass B (layer 1): denom + weighted message scatter ----
__global__ void gat_eacc1(const int* __restrict__ src, const int* __restrict__ dst,
                          const float* __restrict__ as, const float* __restrict__ ad,
                          const float* __restrict__ m, const float* __restrict__ h,
                          float* __restrict__ den, float* __restrict__ agg, int E, int N) {
  int i = blockIdx.x * blockDim.x + threadIdx.x;
  int tot = (E + N) * 8;
  if (i >= tot) return;
  int e = i >> 3, hh = i & 7;
  int s_ = (e < E) ? src[e] : (e - E);
  int d_ = (e < E) ? dst[e] : (e - E);
  float t = lrelu(as[s_ * 8 + hh] + ad[d_ * 8 + hh]);
  float w = __expf(t - m[d_ * 8 + hh]);
  atomicAdd(&den[d_ * 8 + hh], w);
  const float* hs = h + (size_t)s_ * 64 + hh * 8;
  float* ag = agg + (size_t)d_ * 64 + hh * 8;
#pragma unroll
  for (int f = 0; f < 8; ++f) atomicAdd(&ag[f], hs[f] * w);
}

// ---------------- normalize + bias + ELU (in place on agg) ------------------
__global__ void gat_norm1(float* __restrict__ agg, const float* __restrict__ den,
                          const float* __restrict__ b, int N64) {
  int i = blockIdx.x * blockDim.x + threadIdx.x;
  if (i >= N64) return;
  int n = i >> 6, c = i & 63, hh = c >> 3;
  float v = agg[i] / den[n * 8 + hh] + b[c];
  agg[i] = v > 0.f ? v : (__expf(v) - 1.f);          // ELU(alpha=1)
}

// ---------------- Layer 2 GEMM: h2[N,16] = h1a[N,64] @ W2[64,16] (f32 WMMA) --
__global__ void gat_gemm2(const float* __restrict__ h1a, const float* __restrict__ W,
                          float* __restrict__ h2, int Nn) {
  int wave = (int)((blockIdx.x * blockDim.x + threadIdx.x) >> 5);
  int lane = threadIdx.x & 31;
  int tiles = Nn >> 4;
  if (wave >= tiles) return;
  int row   = wave * 16 + (lane & 15);
  int col   = lane & 15;
  int khalf = (lane >> 4) * 2;
  const float* hr = h1a + (size_t)row * 64;
  v8f acc = {};
#pragma unroll
  for (int k0 = 0; k0 < 64; k0 += 4) {
    v2f a, b;
    a.x = hr[k0 + khalf];
    a.y = hr[k0 + khalf + 1];
    b.x = W[(k0 + khalf) * 16 + col];
    b.y = W[(k0 + khalf + 1) * 16 + col];
    acc = __builtin_amdgcn_wmma_f32_16x16x4_f32(false, a, false, b, (short)0, acc,
                                                false, false);
  }
  int rbase = wave * 16 + ((lane & 16) ? 8 : 0);
#pragma unroll
  for (int j = 0; j < 8; ++j)
    h2[(size_t)(rbase + j) * 16 + col] = acc[j];
}

// ---------------- layer-2 attention logits: per node -------------------------
__global__ void gat_alpha2(const float* __restrict__ h2, const float* __restrict__ a_src,
                           const float* __restrict__ a_dst, float* __restrict__ as,
                           float* __restrict__ ad, int N) {
  int n = blockIdx.x * blockDim.x + threadIdx.x;
  if (n >= N) return;
  const float* hp = h2 + (size_t)n * 16;
  float vs = 0.f, vd = 0.f;
#pragma unroll
  for (int c = 0; c < 16; ++c) { float hv = hp[c]; vs += hv * a_src[c]; vd += hv * a_dst[c]; }
  as[n] = vs; ad[n] = vd;
}

__global__ void gat_emax2(const int* __restrict__ src, const int* __restrict__ dst,
                          const float* __restrict__ as, const float* __restrict__ ad,
                          float* __restrict__ m, int E, int N) {
  int e = blockIdx.x * blockDim.x + threadIdx.x;
  if (e >= E + N) return;
  int s_ = (e < E) ? src[e] : (e - E);
  int d_ = (e < E) ? dst[e] : (e - E);
  atomicMaxF(&m[d_], lrelu(as[s_] + ad[d_]));
}

__global__ void gat_eacc2(const int* __restrict__ src, const int* __restrict__ dst,
                          const float* __restrict__ as, const float* __restrict__ ad,
                          const float* __restrict__ m, const float* __restrict__ h2,
                          float* __restrict__ den, float* __restrict__ agg, int E, int N) {
  int i = blockIdx.x * blockDim.x + threadIdx.x;
  int tot = (E + N) * 16;
  if (i >= tot) return;
  int e = i >> 4, c = i & 15;
  int s_ = (e < E) ? src[e] : (e - E);
  int d_ = (e < E) ? dst[e] : (e - E);
  float t = lrelu(as[s_] + ad[d_]);
  float w = __expf(t - m[d_]);
  atomicAdd(&agg[(size_t)d_ * 16 + c], h2[(size_t)s_ * 16 + c] * w);
  if (c == 0) atomicAdd(&den[d_], w);
}

// ---------------- normalize + bias + log_softmax -----------------------------
__global__ void gat_final(const float* __restrict__ agg, const float* __restrict__ den,
                          const float* __restrict__ b, float* __restrict__ out, int N) {
  int n = blockIdx.x * blockDim.x + threadIdx.x;
  if (n >= N) return;
  float v[16];
  float dn = den[n];
  float mx = -3.0e38f;
#pragma unroll
  for (int c = 0; c < 16; ++c) {
    v[c] = agg[(size_t)n * 16 + c] / dn + b[c];
    mx = fmaxf(mx, v[c]);
  }
  float s = 0.f;
#pragma unroll
  for (int c = 0; c < 16; ++c) s += __expf(v[c] - mx);
  float lse = mx + __logf(s);
#pragma unroll
  for (int c = 0; c < 16; ++c) out[(size_t)n * 16 + c] = v[c] - lse;
}

extern "C" void kernel_launch(void* const* d_in, const int* in_sizes, int n_in,
                              void* d_out, int out_size, void* d_ws, size_t ws_size,
                              hipStream_t stream) {
  const float* x      = (const float*)d_in[0];
  const int*   ei     = (const int*)d_in[1];
  const float* W1     = (const float*)d_in[2];
  const float* a_src1 = (const float*)d_in[3];
  const float* a_dst1 = (const float*)d_in[4];
  const float* b1     = (const float*)d_in[5];
  const float* W2     = (const float*)d_in[6];
  const float* a_src2 = (const float*)d_in[7];
  const float* a_dst2 = (const float*)d_in[8];
  const float* b2     = (const float*)d_in[9];
  float* out = (float*)d_out;

  const int N = in_sizes[0] / 512;     // 50000 (multiple of 16)
  const int E = in_sizes[1] / 2;       // 1600000
  const int* src = ei;                 // edge_index[0]
  const int* dst = ei + E;             // edge_index[1]

  // Workspace layout (floats)
  float* ws   = (float*)d_ws;
  float* h1   = ws;                    // N*64
  float* as1  = h1  + (size_t)N * 64;  // N*8
  float* ad1  = as1 + (size_t)N * 8;   // N*8
  float* m1   = ad1 + (size_t)N * 8;   // N*8
  float* den1 = m1  + (size_t)N * 8;   // N*8
  float* agg1 = den1+ (size_t)N * 8;   // N*64  (becomes h1 activated, in place)
  float* h2   = agg1+ (size_t)N * 64;  // N*16
  float* as2  = h2  + (size_t)N * 16;  // N
  float* ad2  = as2 + (size_t)N;       // N
  float* m2   = ad2 + (size_t)N;       // N
  float* den2 = m2  + (size_t)N;       // N
  float* agg2 = den2+ (size_t)N;       // N*16

  const int B = 256;
  const float NEG_INF = -__builtin_huge_valf();
  auto cdiv = [](long long a, long long b) { return (unsigned)((a + b - 1) / b); };

  // --- init accumulators (fresh every call; graph-capture safe) ---
  hipMemsetAsync(den1, 0, (size_t)N * (8 + 64) * sizeof(float), stream); // den1+agg1
  hipMemsetAsync(den2, 0, (size_t)N * (1 + 16) * sizeof(float), stream); // den2+agg2
  gat_fill<<<cdiv((long long)N * 8, B), B, 0, stream>>>(m1, NEG_INF, N * 8);
  gat_fill<<<cdiv(N, B), B, 0, stream>>>(m2, NEG_INF, N);

  // --- layer 1 ---
  int tiles1 = (N / 16) * 4;
  gat_gemm1<<<cdiv((long long)tiles1 * 32, B), B, 0, stream>>>(x, W1, h1, N);
  gat_alpha1<<<cdiv((long long)N * 8, B), B, 0, stream>>>(h1, a_src1, a_dst1, as1, ad1, N * 8);
  gat_emax1<<<cdiv((long long)(E + N) * 8, B), B, 0, stream>>>(src, dst, as1, ad1, m1, E, N);
  gat_eacc1<<<cdiv((long long)(E + N) * 8, B), B, 0, stream>>>(src, dst, as1, ad1, m1, h1,
                                                              den1, agg1, E, N);
  gat_norm1<<<cdiv((long long)N * 64, B), B, 0, stream>>>(agg1, den1, b1, N * 64);

  // --- layer 2 ---
  int tiles2 = N / 16;
  gat_gemm2<<<cdiv((long long)tiles2 * 32, B), B, 0, stream>>>(agg1, W2, h2, N);
  gat_alpha2<<<cdiv(N, B), B, 0, stream>>>(h2, a_src2, a_dst2, as2, ad2, N);
  gat_emax2<<<cdiv(E + N, B), B, 0, stream>>>(src, dst, as2, ad2, m2, E, N);
  gat_eacc2<<<cdiv((long long)(E + N) * 16, B), B, 0, stream>>>(src, dst, as2, ad2, m2, h2,
                                                               den2, agg2, E, N);
  gat_final<<<cdiv(N, B), B, 0, stream>>>(agg2, den2, b2, out, N);
}